// HTGN_32847909879994
// MI455X (gfx1250) — compile-verified
//
#include <hip/hip_runtime.h>
#include <hip/hip_bf16.h>
#include <math.h>
#include <stdint.h>

// ---------------------------------------------------------------------------
// HTGN forward for MI455X (gfx1250, wave32).
// Memory-bound workload (~1.5-2 GB traffic vs ~16 GFLOP of GEMM) -> keep f32
// precision; all GEMMs on V_WMMA_F32_16X16X4_F32; weight panels staged into
// LDS via the Tensor Data Mover (tensor_load_to_lds) where layout permits.
// ---------------------------------------------------------------------------

typedef __attribute__((ext_vector_type(2))) float    v2f;
typedef __attribute__((ext_vector_type(8))) float    v8f;
typedef __attribute__((ext_vector_type(4))) unsigned u32x4;
typedef __attribute__((ext_vector_type(4))) int      i32x4;
typedef __attribute__((ext_vector_type(8))) int      i32x8;

#define MIN_NORM 1e-15f
#define MAXNORM  0.996f      // (1 - 4e-3) / sqrt(c), c == 1
#define LRELU_S  0.01f

#if defined(__has_builtin)
#if __has_builtin(__builtin_amdgcn_tensor_load_to_lds) && __has_builtin(__builtin_amdgcn_s_wait_tensorcnt)
#define HAVE_TDM 1
#endif
#endif
#ifndef HAVE_TDM
#define HAVE_TDM 0
#endif

__device__ __forceinline__ float artanh_f(float x) {
  x = fminf(fmaxf(x, -1.f + 1e-7f), 1.f - 1e-7f);
  return 0.5f * (log1pf(x) - log1pf(-x));
}
__device__ __forceinline__ float sigmoid_f(float x) { return 1.f / (1.f + expf(-x)); }

// ---------------------------------------------------------------------------
// WMMA GEMM: out[nrows, NT*16] = X[nrows, KDIM] @ op(W) (+ bias)
//   WT=true : W stored [NT*16, KDIM] row-major, compute X @ W.T
//             -> B panel is a plain row-major 2D tile: stage via TDM
//                (tensor_load_to_lds, LDS pitch KDIM+2 dwords via pad fields)
//   WT=false: W stored [KDIM, NT*16] row-major, compute X @ W (transposed
//             cooperative stage; TDM cannot transpose)
//   ATT=true: epilogue computes e[row] = sum_n tanh(acc[row][n]) * bias[n]
// B panel k-major in LDS: Bs[n*KP + k]; each lane's (b.x,b.y) is one aligned
// ds_load_b64, landing directly in an even VGPR pair for WMMA.
// nrows must be a multiple of 16 (true for all call sites here).
// ---------------------------------------------------------------------------
template<int KDIM, int NT, bool WT, bool BIAS, bool ATT>
__global__ __launch_bounds__(256)
void gemm_wmma(const float* __restrict__ X, const float* __restrict__ Wm,
               const float* __restrict__ bias, float* __restrict__ out, int nrows)
{
  constexpr int NC = NT * 16;
  constexpr int KP = KDIM + 2;               // padded LDS pitch (dwords), even
  __shared__ __align__(16) float Bs[NC * KP];

#if HAVE_TDM
  if (WT) {
    if (threadIdx.x == 0) {
      // --- Tensor DMA descriptor (D#) for a [NC x KDIM] f32 tile copy ---
      const unsigned lds = (unsigned)(uintptr_t)(&Bs[0]);
      const unsigned long long ga = (unsigned long long)(uintptr_t)Wm;
      u32x4 g0;
      g0.x = 1u;                                              // count=1 valid
      g0.y = lds;                                             // lds_addr (bytes)
      g0.z = (unsigned)(ga & 0xffffffffu);                    // global_addr lo
      g0.w = (unsigned)((ga >> 32) & 0x01ffffffu) | (2u << 30); // addr hi | type=2
      // pad_interval: row = KDIM dwords = KDIM/2 qwords = 2^padi -> padi
      constexpr unsigned padi = (KDIM == 128) ? 6u : 5u;
      i32x8 g1;
      g1[0] = (int)((2u << 16)      // data_size = 4B
                  | (1u << 20)      // pad_enable
                  | (padi << 22)    // pad_interval
                  | (1u << 25));    // pad_amount = 2 dwords -> pitch KDIM+2
      g1[1] = (int)(((unsigned)KDIM & 0xffffu) << 16);        // tensor_dim0 lo16
      g1[2] = (int)((((unsigned)KDIM >> 16) & 0xffffu)        // tensor_dim0 hi16
                  | (((unsigned)NC & 0xffffu) << 16));        // tensor_dim1 lo16
      g1[3] = (int)((((unsigned)NC >> 16) & 0xffffu)          // tensor_dim1 hi16
                  | (((unsigned)KDIM & 0xffffu) << 16));      // tile_dim0
      g1[4] = (int)((unsigned)NC & 0xffffu);                  // tile_dim1 (dim2=0)
      g1[5] = (int)KDIM;                                      // tensor_dim0_stride lo
      g1[6] = 0;
      g1[7] = 0;
      i32x4 z4 = {0, 0, 0, 0};
#if __clang_major__ >= 23
      i32x8 z8 = {0, 0, 0, 0, 0, 0, 0, 0};
      __builtin_amdgcn_tensor_load_to_lds(g0, g1, z4, z4, z8, 0);
#else
      __builtin_amdgcn_tensor_load_to_lds(g0, g1, z4, z4, 0);
#endif
      __builtin_amdgcn_s_wait_tensorcnt(0);
    }
  } else
#endif
  {
    // cooperative stage (and WT=false transpose path): Bs[n*KP + k] = B[k][n]
    for (int i = threadIdx.x; i < KDIM * NC; i += 256) {
      if (WT) { int n = i / KDIM, k = i - n * KDIM; Bs[n * KP + k] = Wm[n * KDIM + k]; }
      else    { int k = i / NC,   n = i - k * NC;   Bs[n * KP + k] = Wm[k * NC + n]; }
    }
  }
  __syncthreads();

  const int wave = threadIdx.x >> 5;
  const int lane = threadIdx.x & 31;
  const int rowBase = (blockIdx.x * 8 + wave) * 16;
  if (rowBase >= nrows) return;

  const int ln   = lane & 15;
  const int hf   = lane >> 4;
  const int koff = hf * 2;

  int mrow = rowBase + ln;
  if (mrow >= nrows) mrow = nrows - 1;       // belt-and-braces (rows 16-aligned)
  const float* xrow = X + (size_t)mrow * KDIM;
  __builtin_prefetch(xrow, 0, 0);

  const v8f zero = {0.f, 0.f, 0.f, 0.f, 0.f, 0.f, 0.f, 0.f};
  v8f acc[NT];
#pragma unroll
  for (int t = 0; t < NT; ++t) acc[t] = zero;

  for (int k0 = 0; k0 < KDIM; k0 += 4) {
    const v2f a = *(const v2f*)(xrow + k0 + koff);            // 8B aligned
#pragma unroll
    for (int t = 0; t < NT; ++t) {
      const v2f b = *(const v2f*)(&Bs[(t * 16 + ln) * KP + k0 + koff]);
      acc[t] = __builtin_amdgcn_wmma_f32_16x16x4_f32(
          false, a, false, b, (short)0, acc[t], false, false);
    }
  }

  if (ATT) {
    // fused: e = tanh(acc_row) . r_att  (bias == r_att here)
#pragma unroll
    for (int r = 0; r < 8; ++r) {
      float p = 0.f;
#pragma unroll
      for (int t = 0; t < NT; ++t) p += tanhf(acc[t][r]) * bias[t * 16 + ln];
      p += __shfl_xor(p, 1);
      p += __shfl_xor(p, 2);
      p += __shfl_xor(p, 4);
      p += __shfl_xor(p, 8);                 // reduce within 16-lane half
      if (ln == 0) out[rowBase + r + hf * 8] = p;
    }
  } else {
#pragma unroll
    for (int t = 0; t < NT; ++t) {
#pragma unroll
      for (int r = 0; r < 8; ++r) {
        const int m = rowBase + r + hf * 8;
        const int n = t * 16 + ln;
        float v = acc[t][r];
        if (BIAS) v += bias[n];
        out[(size_t)m * NC + n] = v;         // no per-element guard: rows %16==0
      }
    }
  }
}

// ---------------------------------------------------------------------------
// Elementwise / graph kernels (one thread per row unless noted)
// ---------------------------------------------------------------------------
__global__ void k_fill(float* p, float v, long long n) {
  long long i = blockIdx.x * (long long)blockDim.x + threadIdx.x;
  if (i < n) p[i] = v;
}

__global__ void k_deg(const int* __restrict__ src, float* __restrict__ deg, int E) {
  int e = blockIdx.x * blockDim.x + threadIdx.x;
  if (e < E) __hip_atomic_fetch_add(&deg[src[e]], 1.0f, __ATOMIC_RELAXED, __HIP_MEMORY_SCOPE_AGENT);
}

__global__ void k_dis(float* deg, int N) {   // in place deg -> 1/sqrt(deg)
  int i = blockIdx.x * blockDim.x + threadIdx.x;
  if (i < N) deg[i] = rsqrtf(fmaxf(deg[i], 1e-12f));
}

// hyperbolic bias: hb = proj(expmap0(b)); hb[D] = ||hb||^2  (single block)
template<int D>
__global__ void k_hb(const float* __restrict__ b, float* __restrict__ hb) {
  int d = threadIdx.x;
  if (d >= D) return;
  float ss = 0.f;
  for (int j = 0; j < D; ++j) ss += b[j] * b[j];
  float un = fmaxf(sqrtf(ss), MIN_NORM);
  float t  = tanhf(un);
  float s  = t / un;
  float ps = (t > MAXNORM) ? (MAXNORM / t) : 1.f;
  hb[d] = b[d] * s * ps;
  if (d == 0) { float fn = fminf(t, MAXNORM); hb[D] = fn * fn; }
}

// x0 = proj(expmap0(tlin)); xcat = proj([x0, hid_last])
__global__ void k_exp_cat(const float* __restrict__ tlin, const float* __restrict__ hid,
                          float* __restrict__ xcat, int N) {
  int i = blockIdx.x * blockDim.x + threadIdx.x;
  if (i >= N) return;
  const float* u = tlin + (size_t)i * 64;
  float y[64];
  float ss = 0.f;
  for (int d = 0; d < 64; ++d) { y[d] = u[d]; ss += y[d] * y[d]; }
  float un = fmaxf(sqrtf(ss), MIN_NORM);
  float t  = tanhf(un);
  float s  = (t / un) * ((t > MAXNORM) ? (MAXNORM / t) : 1.f);
  const float* h = hid + (size_t)i * 64;
  float ss2 = 0.f;
  for (int d = 0; d < 64; ++d) { y[d] *= s; ss2 += y[d] * y[d]; }
  for (int d = 0; d < 64; ++d) { float v = h[d]; ss2 += v * v; }
  float n2 = fmaxf(sqrtf(ss2), MIN_NORM);
  float p2 = (n2 > MAXNORM) ? (MAXNORM / n2) : 1.f;
  float* o = xcat + (size_t)i * 128;
  for (int d = 0; d < 64; ++d) o[d] = y[d] * p2;
  for (int d = 0; d < 64; ++d) o[64 + d] = h[d] * p2;
}

// fused: mv = proj(mobius_matvec) ; out = proj(mobius_add(mv, hb))  (in place on mx)
template<int DIN, int DOUT>
__global__ void k_hyp_post(const float* __restrict__ xin, float* mx,
                           const float* __restrict__ hb, int N) {
  int i = blockIdx.x * blockDim.x + threadIdx.x;
  if (i >= N) return;
  const float* xr = xin + (size_t)i * DIN;
  float ssx = 0.f;
  for (int d = 0; d < DIN; ++d) { float v = xr[d]; ssx += v * v; }
  float xn = fmaxf(sqrtf(ssx), MIN_NORM);
  float* mr = mx + (size_t)i * DOUT;
  float ssm = 0.f, dmb = 0.f;
  for (int d = 0; d < DOUT; ++d) { float v = mr[d]; ssm += v * v; dmb += v * hb[d]; }
  float mxn = fmaxf(sqrtf(ssm), MIN_NORM);
  float smv = (ssm == 0.f) ? 0.f : tanhf((mxn / xn) * artanh_f(xn)) / mxn;
  float mvn = smv * mxn;                                   // ||mv|| before proj
  if (mvn > MAXNORM) smv *= MAXNORM / mvn;
  float x2 = smv * smv * ssm;
  float xy = smv * dmb;
  float y2 = hb[DOUT];
  float cx = 1.f + 2.f * xy + y2;
  float cy = 1.f - x2;
  float den = fmaxf(1.f + 2.f * xy + x2 * y2, MIN_NORM);
  float on2 = fmaxf(cx * cx * x2 + 2.f * cx * cy * xy + cy * cy * y2, 0.f);
  float on  = fmaxf(sqrtf(on2), MIN_NORM) / den;           // ||num/den||
  float pr  = (on > MAXNORM) ? (MAXNORM / on) : 1.f;
  float inv = pr / den;
  for (int d = 0; d < DOUT; ++d) mr[d] = (cx * smv * mr[d] + cy * hb[d]) * inv;
}

// logmap0 (in place allowed: single owner thread per row)
template<int D>
__global__ void k_log0(const float* in, float* out, int nrows) {
  int i = blockIdx.x * blockDim.x + threadIdx.x;
  if (i >= nrows) return;
  const float* r = in + (size_t)i * D;
  float* o = out + (size_t)i * D;
  float ss = 0.f;
  for (int d = 0; d < D; ++d) { float v = r[d]; ss += v * v; }
  float pn = fmaxf(sqrtf(ss), MIN_NORM);
  float s = artanh_f(pn) / pn;
  for (int d = 0; d < D; ++d) o[d] = r[d] * s;
}

// segment-sum scatter: agg[row] += dis[row]*dis[col] * xt[col]; self-loops appended
template<int D>
__global__ void k_scatter(const int* __restrict__ src, const int* __restrict__ dst,
                          const float* __restrict__ dis, const float* __restrict__ xt,
                          float* __restrict__ agg, int E, int N) {
  constexpr int CH = D / 4;
  long long tid = blockIdx.x * (long long)blockDim.x + threadIdx.x;
  long long total = (long long)(E + N) * CH;
  if (tid >= total) return;
  int e = (int)(tid / CH);
  int q = (int)(tid - (long long)e * CH);
  int r, c;
  if (e < E) { r = src[e]; c = dst[e]; } else { r = c = e - E; }
  float w = dis[r] * dis[c];
  const float* xc = xt + (size_t)c * D + q * 4;
  float* ar = agg + (size_t)r * D + q * 4;
#pragma unroll
  for (int j = 0; j < 4; ++j)
    __hip_atomic_fetch_add(&ar[j], w * xc[j], __ATOMIC_RELAXED, __HIP_MEMORY_SCOPE_AGENT);
}

// fused tail of hyp_agg + HypAct:
//   h  = proj(expmap0(agg)); ht = lrelu(logmap0(h));
//   TANGENT ? out = logmap0(proj(expmap0(ht))) : out = proj(expmap0(ht))
template<int D, bool TANGENT>
__global__ void k_agg_post(float* a, int N) {
  int i = blockIdx.x * blockDim.x + threadIdx.x;
  if (i >= N) return;
  float* r = a + (size_t)i * D;
  float ss = 0.f;
  for (int d = 0; d < D; ++d) { float v = r[d]; ss += v * v; }
  float an = fmaxf(sqrtf(ss), MIN_NORM);
  float pn = fminf(tanhf(an), MAXNORM);
  float s  = artanh_f(pn) / an;                 // expmap0+proj+logmap0 collapsed
  float ss2 = 0.f;
  for (int d = 0; d < D; ++d) {
    float v = r[d] * s; v = (v > 0.f) ? v : LRELU_S * v; ss2 += v * v;
  }
  float hn  = fmaxf(sqrtf(ss2), MIN_NORM);
  float pn2 = fminf(tanhf(hn), MAXNORM);
  float fs  = TANGENT ? (artanh_f(pn2) / hn) : (pn2 / hn);
  for (int d = 0; d < D; ++d) {
    float v = r[d] * s; v = (v > 0.f) ? v : LRELU_S * v; r[d] = v * fs;
  }
}

// softmax over window + weighted mean of hlog
__global__ void k_att_h(const float* __restrict__ e, const float* __restrict__ hlog,
                        float* __restrict__ h, int N, int win) {
  int i = blockIdx.x * blockDim.x + threadIdx.x;
  if (i >= N) return;
  float ev[8];
  float m = -1e30f;
  for (int w = 0; w < win; ++w) { ev[w] = e[(size_t)w * N + i]; m = fmaxf(m, ev[w]); }
  float sum = 0.f;
  for (int w = 0; w < win; ++w) { ev[w] = expf(ev[w] - m); sum += ev[w]; }
  float inv = 1.f / (sum * (float)win);         // softmax then mean over win
  for (int w = 0; w < win; ++w) ev[w] *= inv;
  for (int d = 0; d < 64; ++d) {
    float acc = 0.f;
    for (int w = 0; w < win; ++w) acc += ev[w] * hlog[((size_t)w * N + i) * 64 + d];
    h[(size_t)i * 64 + d] = acc;
  }
}

// GRU cell + proj(expmap0(.)) epilogue -> final output
__global__ void k_gru_out(const float* __restrict__ gi, const float* __restrict__ gh,
                          const float* __restrict__ hatt, float* __restrict__ out, int N) {
  int i = blockIdx.x * blockDim.x + threadIdx.x;
  if (i >= N) return;
  const float* gir = gi + (size_t)i * 192;
  const float* ghr = gh + (size_t)i * 192;
  const float* hr  = hatt + (size_t)i * 64;
  float z[64];
  float ss = 0.f;
  for (int d = 0; d < 64; ++d) {
    float rg = sigmoid_f(gir[d] + ghr[d]);
    float zg = sigmoid_f(gir[64 + d] + ghr[64 + d]);
    float ng = tanhf(gir[128 + d] + rg * ghr[128 + d]);
    float v  = (1.f - zg) * ng + zg * hr[d];
    z[d] = v; ss += v * v;
  }
  float un = fmaxf(sqrtf(ss), MIN_NORM);
  float t  = tanhf(un);
  float s  = (t / un) * ((t > MAXNORM) ? (MAXNORM / t) : 1.f);
  for (int d = 0; d < 64; ++d) out[(size_t)i * 64 + d] = z[d] * s;
}

// ---------------------------------------------------------------------------
static inline unsigned cdiv_u(long long a, long long b) { return (unsigned)((a + b - 1) / b); }

extern "C" void kernel_launch(void* const* d_in, const int* in_sizes, int n_in,
                              void* d_out, int out_size, void* d_ws, size_t ws_size,
                              hipStream_t stream) {
  const int*   eidx = (const int*)d_in[0];
  const float* feat = (const float*)d_in[1];
  const float* Wl   = (const float*)d_in[2];
  const float* bl   = (const float*)d_in[3];
  const float* W1   = (const float*)d_in[4];
  const float* b1   = (const float*)d_in[5];
  const float* W2   = (const float*)d_in[6];
  const float* b2   = (const float*)d_in[7];
  const float* Q    = (const float*)d_in[8];
  const float* ratt = (const float*)d_in[9];
  const float* Wih  = (const float*)d_in[10];
  const float* Whh  = (const float*)d_in[11];
  const float* bih  = (const float*)d_in[12];
  const float* bhh  = (const float*)d_in[13];
  const float* hidn = (const float*)d_in[14];

  const long long N = in_sizes[1] / 128;     // feat: [N,128]
  const int  E   = in_sizes[0] / 2;          // edge_index: [2,E]
  const int  WIN = (int)(in_sizes[14] / (N * 64));
  const int* src = eidx;
  const int* dst = eidx + E;

  float* ws = (float*)d_ws;
  // overlap-allocated workspace (lifetimes tracked per phase):
  float* xcat = ws;                           // [N,128]  phase 1-2
  float* buf1 = ws + 128 * N;                 // [N,128]  mx1/h1/xt1 ; then mx2/h2/xt2
  float* buf2 = ws + 256 * N;                 // [N,128]  agg1/x1
  float* hlog = ws;                           // [WIN*N,64] (reuses xcat..x1 after conv2)
  float* gi   = ws;                           // [N,192]  (after hlog consumed)
  float* gh   = ws + 192 * N;                 // [N,192]
  float* xt   = ws + 384 * N;                 // [N,64]   final tangent features
  float* tlin = ws + 448 * N;                 // [N,64]   linear out; later reused as hatt
  float* hatt = ws + 448 * N;                 // [N,64]
  float* dis  = ws + 512 * N;                 // [N]
  float* evec = ws + 513 * N;                 // [WIN*N]
  float* hb1  = ws + (513 + WIN) * N;         // [129]
  float* hb2  = hb1 + 136;                    // [65]

  const int TB = 256;
  const unsigned gN = cdiv_u(N, TB);
  const int iN = (int)N;

  // --- degree / GCN norm ---
  k_fill<<<gN, TB, 0, stream>>>(dis, 1.0f, N);                       // self loops
  k_deg<<<cdiv_u(E, TB), TB, 0, stream>>>(src, dis, E);
  k_dis<<<gN, TB, 0, stream>>>(dis, iN);

  // --- hyperbolic biases ---
  k_hb<128><<<1, 128, 0, stream>>>(b1, hb1);
  k_hb<64><<<1, 64, 0, stream>>>(b2, hb2);

  // --- initial linear + toHyperX + concat hidden ---
  gemm_wmma<128, 4, true, true, false><<<cdiv_u(N, 128), TB, 0, stream>>>(feat, Wl, bl, tlin, iN);
  k_exp_cat<<<gN, TB, 0, stream>>>(tlin, hidn + (size_t)(WIN - 1) * N * 64, xcat, iN);

  // --- HGCN conv 1 (128 -> 128) ---
  gemm_wmma<128, 8, true, false, false><<<cdiv_u(N, 128), TB, 0, stream>>>(xcat, W1, nullptr, buf1, iN);
  k_hyp_post<128, 128><<<gN, TB, 0, stream>>>(xcat, buf1, hb1, iN);  // buf1 = h1
  k_log0<128><<<gN, TB, 0, stream>>>(buf1, buf1, iN);                // buf1 = xt1
  k_fill<<<cdiv_u(128 * N, TB), TB, 0, stream>>>(buf2, 0.f, 128 * N);
  k_scatter<128><<<cdiv_u((long long)(E + N) * 32, TB), TB, 0, stream>>>(src, dst, dis, buf1, buf2, E, iN);
  k_agg_post<128, false><<<gN, TB, 0, stream>>>(buf2, iN);           // buf2 = x1

  // --- HGCN conv 2 (128 -> 64), ends in tangent space ---
  gemm_wmma<128, 4, true, false, false><<<cdiv_u(N, 128), TB, 0, stream>>>(buf2, W2, nullptr, buf1, iN);
  k_hyp_post<128, 64><<<gN, TB, 0, stream>>>(buf2, buf1, hb2, iN);   // buf1 = h2
  k_log0<64><<<gN, TB, 0, stream>>>(buf1, buf1, iN);                 // buf1 = xt2
  k_fill<<<cdiv_u(64 * N, TB), TB, 0, stream>>>(xt, 0.f, 64 * N);
  k_scatter<64><<<cdiv_u((long long)(E + N) * 16, TB), TB, 0, stream>>>(src, dst, dis, buf1, xt, E, iN);
  k_agg_post<64, true><<<gN, TB, 0, stream>>>(xt, iN);               // xt = logmap0(x2)

  // --- HTA window attention ---
  k_log0<64><<<cdiv_u(WIN * N, TB), TB, 0, stream>>>(hidn, hlog, (int)(WIN * N));
  gemm_wmma<64, 4, false, false, true><<<cdiv_u(WIN * N, 128), TB, 0, stream>>>(hlog, Q, ratt, evec, (int)(WIN * N));
  k_att_h<<<gN, TB, 0, stream>>>(evec, hlog, hatt, iN, WIN);

  // --- GRU + toHyperX ---
  gemm_wmma<64, 12, true, true, false><<<cdiv_u(N, 128), TB, 0, stream>>>(xt, Wih, bih, gi, iN);
  gemm_wmma<64, 12, true, true, false><<<cdiv_u(N, 128), TB, 0, stream>>>(hatt, Whh, bhh, gh, iN);
  k_gru_out<<<gN, TB, 0, stream>>>(gi, gh, hatt, (float*)d_out, iN);
}